// CausalGraphReasoning_56624848831214
// MI455X (gfx1250) — compile-verified
//
#include <hip/hip_runtime.h>
#include <hip/hip_bf16.h>
#include <math.h>

typedef __attribute__((ext_vector_type(2))) float v2f;
typedef __attribute__((ext_vector_type(8))) float v8f;

#define N_NODES 64
#define D_IN    128
#define HID     256

// Full-precision CDNA5 WMMA: D(16x16,f32) = A(16x4,f32) x B(4x16,f32) + C
__device__ __forceinline__ v8f wmma4(v2f a, v2f b, v8f c) {
  return __builtin_amdgcn_wmma_f32_16x16x4_f32(
      /*neg_a=*/false, a, /*neg_b=*/false, b,
      /*c_mod=*/(short)0, c, /*reuse_a=*/false, /*reuse_b=*/false);
}

__device__ __forceinline__ float sigmoidf_(float x) {
  return 1.0f / (1.0f + __expf(-x));
}

// ---------------------------------------------------------------------------
// Kernel 1: projections P = x @ Wslice  (fp32 WMMA, M=64 N=256 K=128)
//   p=0: cdW1 rows 0-127   p=1: cdW1 rows 128-255
//   p=2..4: cfW1 rows 0-127 / 128-255 / 256-383
// grid = (4 m-tiles, 5 projections), block = 256 (8 waves).
// W is staged per 32-k chunk into LDS as (W[k][n],W[k+1][n]) float2 pairs so
// every B fragment is a single ds_load_b64; A fragment shared by 2 n-tiles.
// ---------------------------------------------------------------------------
__global__ void proj_kernel(const float* __restrict__ x,
                            const float* __restrict__ cdW1,
                            const float* __restrict__ cfW1,
                            float* __restrict__ ws) {
  const int mt = blockIdx.x;   // 16-row tile of the 64 nodes
  const int p  = blockIdx.y;   // projection id
  const float* W; int r0;
  if (p == 0)      { W = cdW1; r0 = 0;   }
  else if (p == 1) { W = cdW1; r0 = 128; }
  else if (p == 2) { W = cfW1; r0 = 0;   }
  else if (p == 3) { W = cfW1; r0 = 128; }
  else             { W = cfW1; r0 = 256; }
  float* P = ws + (size_t)p * N_NODES * HID;

  __shared__ float  xs[16 * D_IN];      // 8 KB
  __shared__ float2 Wst[16 * HID];      // 32 KB: 16 k-pairs x 256 n
  const int tid = threadIdx.x;
  for (int idx = tid; idx < 16 * D_IN; idx += 256)
    xs[idx] = x[(size_t)(mt * 16 + (idx >> 7)) * D_IN + (idx & 127)];

  const int lane = tid & 31, wave = tid >> 5;
  const int hi = (lane >> 4) & 1;       // 0: lanes 0-15, 1: lanes 16-31
  const int l15 = lane & 15;
  const int n0 = wave * 16;             // first n-tile
  const int n1 = (wave + 8) * 16;       // second n-tile

  v8f c0 = {}, c1 = {};
  for (int kc = 0; kc < D_IN; kc += 32) {
    __syncthreads();                    // prior chunk readers done (also xs fill)
    for (int idx = tid; idx < 16 * HID; idx += 256) {
      const int kp = idx >> 8, n = idx & 255;
      const int kr = r0 + kc + 2 * kp;
      float2 w;
      w.x = W[(size_t)kr * HID + n];
      w.y = W[(size_t)(kr + 1) * HID + n];
      Wst[idx] = w;
    }
    __syncthreads();
#pragma unroll
    for (int k0 = 0; k0 < 32; k0 += 4) {
      const int ka = k0 + 2 * hi;       // lanes 0-15: K0/K1, 16-31: K2/K3
      const float2 av = *(const float2*)&xs[l15 * D_IN + kc + ka];
      v2f a; a.x = av.x; a.y = av.y;
      const int kp = ka >> 1;
      const float2 b0 = Wst[kp * HID + n0 + l15];
      const float2 b1 = Wst[kp * HID + n1 + l15];
      v2f vb0; vb0.x = b0.x; vb0.y = b0.y;
      v2f vb1; vb1.x = b1.x; vb1.y = b1.y;
      c0 = wmma4(a, vb0, c0);
      c1 = wmma4(a, vb1, c1);
    }
  }
#pragma unroll
  for (int r = 0; r < 8; ++r) {         // C/D: VGPR r -> row m0+r(+8), lane -> col
    P[(size_t)(mt * 16 + r + 8 * hi) * HID + n0 + l15] = c0[r];
    P[(size_t)(mt * 16 + r + 8 * hi) * HID + n1 + l15] = c1[r];
  }
}

// ---------------------------------------------------------------------------
// Kernel 2: causal matrix. One block per i, two passes of 32 j's.
//   h1[j,h] = relu(PA[i,h]+PB[j,h]+cdb1[h])      (LDS)
//   h2 = relu(h1 @ cdW2 + cdb2)  via fp32 WMMA (cdW2 pair-staged in LDS),
//   fused: score[j] = sum_n h2[j,n]*cdW3[n]; causal = sigmoid(score+cdb3).
// Wave w owns n-tile w for BOTH m-tiles -> B fragment reused, 2 WMMA chains.
// ---------------------------------------------------------------------------
__global__ void causal_kernel(const float* __restrict__ PA,
                              const float* __restrict__ PB,
                              const float* __restrict__ cdb1,
                              const float* __restrict__ cdW2,
                              const float* __restrict__ cdb2,
                              const float* __restrict__ cdW3,
                              const float* __restrict__ cdb3,
                              float* __restrict__ out_causal) {
  const int i = (int)blockIdx.x;
  __shared__ float  h1[32 * HID];       // 32 KB
  __shared__ float2 Wst[16 * 128];      // 16 KB: 16 k-pairs x 128 n
  __shared__ float  score[N_NODES];
  const int tid = threadIdx.x;
  const int lane = tid & 31, wave = tid >> 5;
  const int hi = (lane >> 4) & 1, l15 = lane & 15;
  const int n0 = wave * 16;
  if (tid < N_NODES) score[tid] = 0.0f;

  for (int pass = 0; pass < 2; ++pass) {
    // Phase A: layer-1 activations for this pass's 32 j's
    for (int idx = tid; idx < 32 * HID; idx += 256) {
      const int jj = idx >> 8, h = idx & 255;
      const int j = pass * 32 + jj;
      h1[idx] = fmaxf(PA[i * HID + h] + PB[j * HID + h] + cdb1[h], 0.0f);
    }
    // Phase B: (32x256)@(256x128) WMMA, K staged in 32-wide chunks
    v8f c0 = {}, c1 = {};
    for (int kc = 0; kc < HID; kc += 32) {
      __syncthreads();                  // phase A done / prior chunk readers done
      for (int idx = tid; idx < 16 * 128; idx += 256) {
        const int kp = idx >> 7, n = idx & 127;
        const int kr = kc + 2 * kp;
        float2 w;
        w.x = cdW2[(size_t)kr * 128 + n];
        w.y = cdW2[(size_t)(kr + 1) * 128 + n];
        Wst[idx] = w;
      }
      __syncthreads();
#pragma unroll
      for (int k0 = 0; k0 < 32; k0 += 4) {
        const int ka = k0 + 2 * hi;
        const int kp = ka >> 1;
        const float2 bv = Wst[kp * 128 + n0 + l15];
        v2f b; b.x = bv.x; b.y = bv.y;
        const float2 a0 = *(const float2*)&h1[l15 * HID + kc + ka];
        const float2 a1 = *(const float2*)&h1[(16 + l15) * HID + kc + ka];
        v2f va0; va0.x = a0.x; va0.y = a0.y;
        v2f va1; va1.x = a1.x; va1.y = a1.y;
        c0 = wmma4(va0, b, c0);         // rows 0-15 of this pass
        c1 = wmma4(va1, b, c1);         // rows 16-31 of this pass
      }
    }
    // Epilogue: relu + dot with cdW3, reduce 16 columns per half-wave
    const int n = n0 + l15;
    const float w3 = cdW3[n];
    const float bb = cdb2[n];
#pragma unroll
    for (int r = 0; r < 8; ++r) {
      float p0 = fmaxf(c0[r] + bb, 0.0f) * w3;
      float p1 = fmaxf(c1[r] + bb, 0.0f) * w3;
      p0 += __shfl_xor(p0, 1, 32); p1 += __shfl_xor(p1, 1, 32);
      p0 += __shfl_xor(p0, 2, 32); p1 += __shfl_xor(p1, 2, 32);
      p0 += __shfl_xor(p0, 4, 32); p1 += __shfl_xor(p1, 4, 32);
      p0 += __shfl_xor(p0, 8, 32); p1 += __shfl_xor(p1, 8, 32);
      if (l15 == 0) {
        atomicAdd(&score[pass * 32 + r + 8 * hi], p0);
        atomicAdd(&score[pass * 32 + 16 + r + 8 * hi], p1);
      }
    }
    __syncthreads();                    // h1/score safe before next pass / read
  }
  if (tid < N_NODES) {
    const float s = score[tid] + cdb3[0];
    out_causal[i * N_NODES + tid] = (tid == i) ? 0.0f : sigmoidf_(s);
  }
}

// ---------------------------------------------------------------------------
// Kernel 3: confounder scores. One block per (i,j); wave w handles k=8w..8w+7.
//   tri[i,j,k] = sigmoid(sum_h relu(QA[i,h]+QB[j,h]+QC[k,h]+cfb1[h])*cfW2[h]+cfb2)
// relu blocks GEMM factorization -> VALU dot + full-wave shfl reduction.
// ---------------------------------------------------------------------------
__global__ void confounder_kernel(const float* __restrict__ QA,
                                  const float* __restrict__ QB,
                                  const float* __restrict__ QC,
                                  const float* __restrict__ cfb1,
                                  const float* __restrict__ cfW2,
                                  const float* __restrict__ cfb2,
                                  float* __restrict__ out_conf) {
  const int j = (int)blockIdx.x, i = (int)blockIdx.y;
  __shared__ float base[HID];
  __shared__ float w2l[HID];
  const int tid = threadIdx.x;          // 256 threads == HID
  base[tid] = QA[i * HID + tid] + QB[j * HID + tid] + cfb1[tid];
  w2l[tid]  = cfW2[tid];
  __syncthreads();

  const int lane = tid & 31, wave = tid >> 5;
  const float b2 = cfb2[0];
  float* orow = out_conf + ((size_t)(i * N_NODES + j)) * N_NODES;
  for (int kk = 0; kk < 8; ++kk) {
    const int k = wave * 8 + kk;
    float s = 0.0f;
#pragma unroll
    for (int u = 0; u < 8; ++u) {
      const int h = lane + u * 32;      // bank-conflict-free, coalesced QC
      s = fmaf(fmaxf(base[h] + QC[(size_t)k * HID + h], 0.0f), w2l[h], s);
    }
    s += __shfl_xor(s, 16, 32);
    s += __shfl_xor(s, 8, 32);
    s += __shfl_xor(s, 4, 32);
    s += __shfl_xor(s, 2, 32);
    s += __shfl_xor(s, 1, 32);
    if (lane == 0) {
      const bool distinct = (i != j) && (j != k) && (i != k);
      orow[k] = distinct ? sigmoidf_(s + b2) : 0.0f;
    }
  }
}

// ---------------------------------------------------------------------------
// Kernel 4a/4b: modified_features = x with intervened row replaced by MLP out
// ---------------------------------------------------------------------------
__global__ void copy_kernel(const float* __restrict__ x, float* __restrict__ out_mod) {
  const int idx = blockIdx.x * 256 + threadIdx.x;
  if (idx < N_NODES * D_IN) out_mod[idx] = x[idx];
}

__global__ void intervention_kernel(const float* __restrict__ x,
                                    const int* __restrict__ node_p,
                                    const float* __restrict__ val_p,
                                    const float* __restrict__ ipW1,
                                    const float* __restrict__ ipb1,
                                    const float* __restrict__ ipW2,
                                    const float* __restrict__ ipb2,
                                    float* __restrict__ out_mod) {
  __shared__ float in_l[D_IN + 1];
  __shared__ float ih[HID];
  const int tid = threadIdx.x;
  const int node = node_p[0];
  if (tid < D_IN) in_l[tid] = x[(size_t)node * D_IN + tid];
  if (tid == D_IN) in_l[D_IN] = val_p[0];
  __syncthreads();
  float s = ipb1[tid];
  for (int c = 0; c <= D_IN; ++c)
    s = fmaf(in_l[c], ipW1[(size_t)c * HID + tid], s);
  ih[tid] = fmaxf(s, 0.0f);
  __syncthreads();
  if (tid < D_IN) {
    float e = ipb2[tid];
    for (int h = 0; h < HID; ++h)
      e = fmaf(ih[h], ipW2[(size_t)h * D_IN + tid], e);
    out_mod[(size_t)node * D_IN + tid] = e;
  }
}

// ---------------------------------------------------------------------------
extern "C" void kernel_launch(void* const* d_in, const int* in_sizes, int n_in,
                              void* d_out, int out_size, void* d_ws, size_t ws_size,
                              hipStream_t stream) {
  (void)in_sizes; (void)n_in; (void)out_size; (void)ws_size;
  const float* x     = (const float*)d_in[0];
  const int*   node  = (const int*)  d_in[1];
  const float* val   = (const float*)d_in[2];
  const float* cdW1  = (const float*)d_in[3];
  const float* cdb1  = (const float*)d_in[4];
  const float* cdW2  = (const float*)d_in[5];
  const float* cdb2  = (const float*)d_in[6];
  const float* cdW3  = (const float*)d_in[7];
  const float* cdb3  = (const float*)d_in[8];
  const float* cfW1  = (const float*)d_in[9];
  const float* cfb1  = (const float*)d_in[10];
  const float* cfW2  = (const float*)d_in[11];
  const float* cfb2  = (const float*)d_in[12];
  const float* ipW1  = (const float*)d_in[13];
  const float* ipb1  = (const float*)d_in[14];
  const float* ipW2  = (const float*)d_in[15];
  const float* ipb2  = (const float*)d_in[16];

  float* out        = (float*)d_out;
  float* out_causal = out;                                   // 64*64
  float* out_conf   = out + N_NODES * N_NODES;               // 64^3
  float* out_mod    = out + N_NODES * N_NODES
                          + N_NODES * N_NODES * N_NODES;     // 64*128

  float* ws = (float*)d_ws;                                  // 5 x 64 x 256 f32
  const size_t PS = (size_t)N_NODES * HID;
  const float* PA = ws + 0 * PS;
  const float* PB = ws + 1 * PS;
  const float* QA = ws + 2 * PS;
  const float* QB = ws + 3 * PS;
  const float* QC = ws + 4 * PS;

  proj_kernel<<<dim3(4, 5), 256, 0, stream>>>(x, cdW1, cfW1, ws);
  causal_kernel<<<N_NODES, 256, 0, stream>>>(PA, PB, cdb1, cdW2, cdb2,
                                             cdW3, cdb3, out_causal);
  confounder_kernel<<<dim3(N_NODES, N_NODES), 256, 0, stream>>>(
      QA, QB, QC, cfb1, cfW2, cfb2, out_conf);
  copy_kernel<<<(N_NODES * D_IN + 255) / 256, 256, 0, stream>>>(x, out_mod);
  intervention_kernel<<<1, 256, 0, stream>>>(x, node, val, ipW1, ipb1,
                                             ipW2, ipb2, out_mod);
}